// IterativeMessagePoolingPassingLayer_58119497450153
// MI455X (gfx1250) — compile-verified
//
#include <hip/hip_runtime.h>
#include <hip/hip_bf16.h>

typedef __attribute__((ext_vector_type(2))) float v2f;
typedef __attribute__((ext_vector_type(8))) float v8f;

__device__ __forceinline__ float sigmoidf_(float x) { return 1.0f / (1.0f + expf(-x)); }

// ---------------------------------------------------------------------------
// WMMA GEMM: C[M,N] = A[M,K] x B[K,N]
//   A: row-major weights (m*K + k)  -- matches (OC,IC,2,2) and (3H,H) layouts
//   BMODE 0: dense B[k*N + n] (feature-major activations)
//   BMODE 1: im2col of NCHW x (inC=128, 7x7 in, 2x2 kern, 6x6 out); n = b*36 + oy*6 + ox
//   BMODE 2: im2col of pooled P1 [1024, batch*9] (3x3 in, 2x2 out);  n = b*4 + oy*2 + ox
//   EPI 0: C[m*N + n] = acc
//   EPI 1: bias+relu, 2x2 maxpool 6x6->3x3 via atomicMax into C[m, batch*9] (pre-zeroed)
//   EPI 2: bias+relu, 2x2 maxpool 2x2->1x1 via atomicMax into C[m, batch]   (pre-zeroed)
// Block = 128 threads (4 waves); macro-tile 128(M) x 64(N); K step 16.
// Each wave owns 32 M-rows (2 x 16) over all 64 N-cols -> 8 v8f accumulators.
// B tile stored K-contiguous in LDS so every fragment is one aligned ds_load_b64.
// All M multiples of 128, N multiples of 64, K multiples of 16 in this workload.
// ---------------------------------------------------------------------------
template <int BMODE, int EPI>
__global__ __launch_bounds__(128) void gemm_wmma(
    const float* __restrict__ A, const float* __restrict__ B,
    const float* __restrict__ bias, float* __restrict__ C,
    int M, int N, int K, int batch)
{
  __shared__ __align__(16) float As[128 * 20];  // [m][k], stride 20 (pad)
  __shared__ __align__(16) float Bs[64 * 18];   // [n][k], stride 18 (pad, even)

  const int tid   = threadIdx.x;
  const int wave  = tid >> 5;
  const int lane  = tid & 31;
  const int half  = lane >> 4;   // 0: lanes 0-15, 1: lanes 16-31
  const int ml    = lane & 15;
  const int tileN = blockIdx.x * 64;
  const int tileM = blockIdx.y * 128;

  v8f zero = {0.f, 0.f, 0.f, 0.f, 0.f, 0.f, 0.f, 0.f};
  v8f acc[8] = {zero, zero, zero, zero, zero, zero, zero, zero};

  for (int k0 = 0; k0 < K; k0 += 16) {
    {   // A tile: 128 rows x 16 cols; thread t loads its whole row (4 x float4)
      const float* src = A + (long)(tileM + tid) * K + k0;
      *(float4*)(&As[tid * 20 + 0])  = *(const float4*)(src);
      *(float4*)(&As[tid * 20 + 4])  = *(const float4*)(src + 4);
      *(float4*)(&As[tid * 20 + 8])  = *(const float4*)(src + 8);
      *(float4*)(&As[tid * 20 + 12]) = *(const float4*)(src + 12);
    }
    {   // B tile: 16 k-rows x 64 n-cols, stored transposed Bs[n][k]
      int kk = tid >> 3;
      int c0 = (tid & 7) * 8;
      int k  = k0 + kk;
      if (BMODE == 0) {
        const float* src = B + (long)k * N + tileN + c0;
        float4 q0 = *(const float4*)(src);
        float4 q1 = *(const float4*)(src + 4);
        Bs[(c0 + 0) * 18 + kk] = q0.x;
        Bs[(c0 + 1) * 18 + kk] = q0.y;
        Bs[(c0 + 2) * 18 + kk] = q0.z;
        Bs[(c0 + 3) * 18 + kk] = q0.w;
        Bs[(c0 + 4) * 18 + kk] = q1.x;
        Bs[(c0 + 5) * 18 + kk] = q1.y;
        Bs[(c0 + 6) * 18 + kk] = q1.z;
        Bs[(c0 + 7) * 18 + kk] = q1.w;
      } else if (BMODE == 1) {
        int ic = k >> 2, kr = k & 3, ky = kr >> 1, kx = kr & 1;
#pragma unroll
        for (int j = 0; j < 8; ++j) {
          int n = tileN + c0 + j;
          int b = n / 36, s = n - b * 36;
          int oy = s / 6, ox = s - oy * 6;
          Bs[(c0 + j) * 18 + kk] =
              B[(((long)b * 128 + ic) * 7 + oy + ky) * 7 + ox + kx];
        }
      } else {
        int ic = k >> 2, kr = k & 3, ky = kr >> 1, kx = kr & 1;
#pragma unroll
        for (int j = 0; j < 8; ++j) {
          int n = tileN + c0 + j;
          int b = n >> 2, s = n & 3;
          int oy = s >> 1, ox = s & 1;
          Bs[(c0 + j) * 18 + kk] =
              B[(long)ic * (batch * 9) + b * 9 + (oy + ky) * 3 + (ox + kx)];
        }
      }
    }
    __syncthreads();

#pragma unroll
    for (int kk = 0; kk < 16; kk += 4) {
      const int ks = kk + half * 2;  // lanes 0-15: K=kk,kk+1 ; lanes 16-31: K=kk+2,kk+3
      v2f a0 = *(const v2f*)(&As[(wave * 32 + ml) * 20 + ks]);
      v2f a1 = *(const v2f*)(&As[(wave * 32 + 16 + ml) * 20 + ks]);
#pragma unroll
      for (int nt = 0; nt < 4; ++nt) {
        v2f b = *(const v2f*)(&Bs[(nt * 16 + ml) * 18 + ks]);
        acc[nt] = __builtin_amdgcn_wmma_f32_16x16x4_f32(
            false, a0, false, b, (short)0, acc[nt], false, false);
        acc[4 + nt] = __builtin_amdgcn_wmma_f32_16x16x4_f32(
            false, a1, false, b, (short)0, acc[4 + nt], false, false);
      }
    }
    __syncthreads();
  }

  // epilogue: VGPR r of acc[mt*?] holds (M = mt*16 + half*8 + r, N = ml) of the tile
#pragma unroll
  for (int mt = 0; mt < 2; ++mt) {
    const int mBase = tileM + wave * 32 + mt * 16 + half * 8;
#pragma unroll
    for (int nt = 0; nt < 4; ++nt) {
      int n = tileN + nt * 16 + ml;
      v8f av = acc[mt * 4 + nt];
#pragma unroll
      for (int r = 0; r < 8; ++r) {
        int m = mBase + r;
        float val = av[r];
        if (EPI == 0) {
          C[(long)m * N + n] = val;
        } else {
          val = fmaxf(val + bias[m], 0.f);  // relu -> val >= 0, so uint atomicMax == float max
          if (EPI == 1) {
            int b = n / 36, s = n - b * 36;
            int oy = s / 6, ox = s - oy * 6;
            atomicMax((unsigned int*)&C[(long)m * (batch * 9) + b * 9 + (oy >> 1) * 3 + (ox >> 1)],
                      __float_as_uint(val));
          } else {
            atomicMax((unsigned int*)&C[(long)m * batch + (n >> 2)], __float_as_uint(val));
          }
        }
      }
    }
  }
}

// ---------------------------------------------------------------------------
// GRU pointwise: h_out[f,r] from gi[3H,R], gh[3H,R] (nullable -> 0), biases, h (nullable -> 0)
// ---------------------------------------------------------------------------
__global__ __launch_bounds__(256) void gru_pointwise(
    const float* __restrict__ gi, const float* __restrict__ gh,
    const float* __restrict__ bih, const float* __restrict__ bhh,
    const float* __restrict__ h, float* __restrict__ hout, int H, int R)
{
  long idx = (long)blockIdx.x * blockDim.x + threadIdx.x;
  if (idx >= (long)H * R) return;
  int f = (int)(idx / R);
  int r = (int)(idx - (long)f * R);

  float ir = gi[(long)f * R + r] + bih[f];
  float iz = gi[(long)(H + f) * R + r] + bih[H + f];
  float in_ = gi[(long)(2 * H + f) * R + r] + bih[2 * H + f];
  float hr = (gh ? gh[(long)f * R + r] : 0.f) + bhh[f];
  float hz = (gh ? gh[(long)(H + f) * R + r] : 0.f) + bhh[H + f];
  float hn = (gh ? gh[(long)(2 * H + f) * R + r] : 0.f) + bhh[2 * H + f];

  float rg = sigmoidf_(ir + hr);
  float z  = sigmoidf_(iz + hz);
  float n  = tanhf(in_ + rg * hn);
  float hp = h ? h[(long)f * R + r] : 0.f;
  hout[(long)f * R + r] = (1.f - z) * n + z * hp;
}

// ---------------------------------------------------------------------------
// Edge gates + message construction. One wave32 per edge.
//   gn = sigmoid([ns,eh]·w_np), gs = sigmoid([ns,eh]·w_es), go = sigmoid([no,eh]·w_eo)
//   emsg[:,g]  = gs*ns + go*no
//   npool[:,subj] += gn*eh ; npool[:,obj] += gn*eh   (== mat @ (gn*eh))
// H fixed at 512 (16 features per lane).
// ---------------------------------------------------------------------------
__global__ __launch_bounds__(128) void edge_gates(
    const float* __restrict__ nh, const float* __restrict__ eh,
    const int* __restrict__ eidx,
    const float* __restrict__ wn, const float* __restrict__ wsu,
    const float* __restrict__ wo,
    float* __restrict__ emsg, float* __restrict__ npool,
    int E, int N, int NodeR, int EdgeR)
{
  int g = blockIdx.x * 4 + (threadIdx.x >> 5);
  if (g >= EdgeR) return;
  int lane = threadIdx.x & 31;
  int b = g / E;
  int subj = eidx[g * 3 + 1];
  int obj  = eidx[g * 3 + 2];
  int cs = b * N + subj;
  int co = b * N + obj;

  float nsv[16], nov[16], ehv[16];
  float dn = 0.f, dsu = 0.f, dob = 0.f;
#pragma unroll
  for (int i = 0; i < 16; ++i) {
    int f = lane + i * 32;
    float a  = nh[(long)f * NodeR + cs];
    float c  = nh[(long)f * NodeR + co];
    float ev = eh[(long)f * EdgeR + g];
    nsv[i] = a; nov[i] = c; ehv[i] = ev;
    dn  += a * wn[f]  + ev * wn[512 + f];
    dsu += a * wsu[f] + ev * wsu[512 + f];
    dob += c * wo[f]  + ev * wo[512 + f];
  }
#pragma unroll
  for (int o = 16; o; o >>= 1) {
    dn  += __shfl_xor(dn, o, 32);
    dsu += __shfl_xor(dsu, o, 32);
    dob += __shfl_xor(dob, o, 32);
  }
  float gn = sigmoidf_(dn), gs = sigmoidf_(dsu), go = sigmoidf_(dob);
#pragma unroll
  for (int i = 0; i < 16; ++i) {
    int f = lane + i * 32;
    emsg[(long)f * EdgeR + g] = gs * nsv[i] + go * nov[i];
    float nm = gn * ehv[i];
    atomicAdd(&npool[(long)f * NodeR + cs], nm);
    atomicAdd(&npool[(long)f * NodeR + co], nm);
  }
}

// ---------------------------------------------------------------------------
// [feat,row] -> row-major (B*N,H) ++ (B*E,H) into d_out
// ---------------------------------------------------------------------------
__global__ __launch_bounds__(256) void transpose_out(
    const float* __restrict__ nh, const float* __restrict__ eh,
    float* __restrict__ out, int H, int Rn, int Re)
{
  long idx = (long)blockIdx.x * blockDim.x + threadIdx.x;
  long nTot = (long)H * Rn;
  long tot = nTot + (long)H * Re;
  if (idx >= tot) return;
  if (idx < nTot) {
    int r = (int)(idx / H);
    int f = (int)(idx - (long)r * H);
    out[idx] = nh[(long)f * Rn + r];
  } else {
    long j = idx - nTot;
    int r = (int)(j / H);
    int f = (int)(j - (long)r * H);
    out[idx] = eh[(long)f * Re + r];
  }
}

extern "C" void kernel_launch(void* const* d_in, const int* in_sizes, int n_in,
                              void* d_out, int out_size, void* d_ws, size_t ws_size,
                              hipStream_t stream)
{
  const float* node_latents = (const float*)d_in[0];
  const float* edge_latents = (const float*)d_in[1];
  const int*   eidx         = (const int*)d_in[2];
  const float* node_cw1 = (const float*)d_in[3];
  const float* node_cb1 = (const float*)d_in[4];
  const float* node_cw2 = (const float*)d_in[5];
  const float* node_cb2 = (const float*)d_in[6];
  const float* edge_cw1 = (const float*)d_in[7];
  const float* edge_cb1 = (const float*)d_in[8];
  const float* edge_cw2 = (const float*)d_in[9];
  const float* edge_cb2 = (const float*)d_in[10];
  const float* n_wih = (const float*)d_in[11];
  const float* n_whh = (const float*)d_in[12];
  const float* n_bih = (const float*)d_in[13];
  const float* n_bhh = (const float*)d_in[14];
  const float* e_wih = (const float*)d_in[15];
  const float* e_whh = (const float*)d_in[16];
  const float* e_bih = (const float*)d_in[17];
  const float* e_bhh = (const float*)d_in[18];
  const float* w_np  = (const float*)d_in[19];
  const float* w_es  = (const float*)d_in[20];
  const float* w_eo  = (const float*)d_in[21];
  (void)in_sizes; (void)n_in; (void)ws_size;

  const int B = 2, N = 32, E = 992, H = 512, EMB = 128;
  const int Rn = B * N;    // 64 node rows
  const int Re = B * E;    // 1984 edge rows

  float* ws = (float*)d_ws;
  size_t off = 0;
  auto alloc = [&](size_t nf) { float* p = ws + off; off += nf; return p; };
  float* P1e    = alloc((size_t)1024 * Re * 9);   // pooled conv1 (edges)  [1024, Re*9]
  float* P1n    = alloc((size_t)1024 * Rn * 9);   // pooled conv1 (nodes)
  float* feat_e = alloc((size_t)H * Re);          // pooled conv2 == encoder features
  float* feat_n = alloc((size_t)H * Rn);
  float* eh     = alloc((size_t)H * Re);          // edge hidden state [H, Re]
  float* nh     = alloc((size_t)H * Rn);          // node hidden state [H, Rn]
  float* emsg   = alloc((size_t)H * Re);
  float* npool  = alloc((size_t)H * Rn);
  // gate buffers alias the (dead after conv2) P1e region: need 2*3H*Re + 2*3H*Rn
  // = 6,291,456 floats < 18,284,544 floats of P1e.
  float* gi_e = P1e;
  float* gh_e = P1e + (size_t)3 * H * Re;
  float* gi_n = P1e + (size_t)6 * H * Re;
  float* gh_n = gi_n + (size_t)3 * H * Rn;

  // ---- CNN encoders ----
  hipMemsetAsync(P1e, 0, (size_t)1024 * Re * 9 * sizeof(float), stream);
  hipMemsetAsync(P1n, 0, (size_t)1024 * Rn * 9 * sizeof(float), stream);
  // conv1: M=1024, K=128*4=512, N=rows*36 (6x6 out), fused bias+relu+pool -> [1024, rows*9]
  gemm_wmma<1, 1><<<dim3(Re * 36 / 64, 1024 / 128), 128, 0, stream>>>(
      edge_cw1, edge_latents, edge_cb1, P1e, 1024, Re * 36, EMB * 4, Re);
  gemm_wmma<1, 1><<<dim3(Rn * 36 / 64, 1024 / 128), 128, 0, stream>>>(
      node_cw1, node_latents, node_cb1, P1n, 1024, Rn * 36, EMB * 4, Rn);

  hipMemsetAsync(feat_e, 0, (size_t)H * Re * sizeof(float), stream);
  hipMemsetAsync(feat_n, 0, (size_t)H * Rn * sizeof(float), stream);
  // conv2: M=512, K=1024*4=4096, N=rows*4 (2x2 out), fused bias+relu+pool -> [512, rows]
  gemm_wmma<2, 2><<<dim3(Re * 4 / 64, H / 128), 128, 0, stream>>>(
      edge_cw2, P1e, edge_cb2, feat_e, H, Re * 4, 4096, Re);
  gemm_wmma<2, 2><<<dim3(Rn * 4 / 64, H / 128), 128, 0, stream>>>(
      node_cw2, P1n, node_cb2, feat_n, H, Rn * 4, 4096, Rn);

  // ---- initial GRU (h = 0 -> skip W_hh GEMM; gh contribution is just b_hh) ----
  gemm_wmma<0, 0><<<dim3(Re / 64, 1536 / 128), 128, 0, stream>>>(
      e_wih, feat_e, nullptr, gi_e, 1536, Re, H, 0);
  gemm_wmma<0, 0><<<dim3(Rn / 64, 1536 / 128), 128, 0, stream>>>(
      n_wih, feat_n, nullptr, gi_n, 1536, Rn, H, 0);
  gru_pointwise<<<(H * Re + 255) / 256, 256, 0, stream>>>(
      gi_e, nullptr, e_bih, e_bhh, nullptr, eh, H, Re);
  gru_pointwise<<<(H * Rn + 255) / 256, 256, 0, stream>>>(
      gi_n, nullptr, n_bih, n_bhh, nullptr, nh, H, Rn);

  // ---- message passing: reference returns state captured entering iteration 3,
  //      i.e. after exactly 2 GRU-update rounds ----
  for (int it = 0; it < 2; ++it) {
    hipMemsetAsync(npool, 0, (size_t)H * Rn * sizeof(float), stream);
    edge_gates<<<Re / 4, 128, 0, stream>>>(
        nh, eh, eidx, w_np, w_es, w_eo, emsg, npool, E, N, Rn, Re);

    // node GRU update
    gemm_wmma<0, 0><<<dim3(Rn / 64, 1536 / 128), 128, 0, stream>>>(
        n_wih, npool, nullptr, gi_n, 1536, Rn, H, 0);
    gemm_wmma<0, 0><<<dim3(Rn / 64, 1536 / 128), 128, 0, stream>>>(
        n_whh, nh, nullptr, gh_n, 1536, Rn, H, 0);
    gru_pointwise<<<(H * Rn + 255) / 256, 256, 0, stream>>>(
        gi_n, gh_n, n_bih, n_bhh, nh, nh, H, Rn);

    // edge GRU update
    gemm_wmma<0, 0><<<dim3(Re / 64, 1536 / 128), 128, 0, stream>>>(
        e_wih, emsg, nullptr, gi_e, 1536, Re, H, 0);
    gemm_wmma<0, 0><<<dim3(Re / 64, 1536 / 128), 128, 0, stream>>>(
        e_whh, eh, nullptr, gh_e, 1536, Re, H, 0);
    gru_pointwise<<<(H * Re + 255) / 256, 256, 0, stream>>>(
        gi_e, gh_e, e_bih, e_bhh, eh, eh, H, Re);
  }

  // ---- write (B,N,H) ++ (B,E,H) row-major ----
  transpose_out<<<(out_size + 255) / 256, 256, 0, stream>>>(
      nh, eh, (float*)d_out, H, Rn, Re);
}